// PointSIFT_module_basic_4389456577473
// MI455X (gfx1250) — compile-verified
//
#include <hip/hip_runtime.h>

typedef float v2f __attribute__((ext_vector_type(2)));
typedef float v8f __attribute__((ext_vector_type(8)));

#define NPTS   4096
#define NB     2
#define CF     64
#define QPB    64          // queries per block = 2 waves * 32
#define NCHUNK (NPTS / 16)

// ---------------------------------------------------------------------------
// Kernel 1: octant-wise nearest-neighbor selection via V_WMMA_F32_16X16X4_F32
// dist^2 tile computed as D = A*B + C with
//   A row m = (-2qx, -2qy, -2qz, |q|^2), B col n = (px, py, pz, 1), C = |p|^2
// ---------------------------------------------------------------------------
__global__ __launch_bounds__(QPB) void pointsift_select_wmma(
    const float* __restrict__ xyz_all,
    const float* __restrict__ radius_p,
    float* __restrict__ idx_out)        // [B][N][8] as float
{
  __shared__ float xs[NPTS], ys[NPTS], zs[NPTS];   // 48 KB: whole batch coords
  __shared__ float dstage[2 * 32 * 16];            // per-wave 32x16 dist tile
  __shared__ float minv[8 * QPB];                  // [oct][tid] running min
  __shared__ int   mina[8 * QPB];                  // [oct][tid] running argmin

  const int tid = threadIdx.x;                     // 0..63
  const int b   = blockIdx.y;
  const float r = radius_p[0];
  const float judge = r * r;

  const float* xyz = xyz_all + (size_t)b * NPTS * 3;

  // cooperative load of all coords for this batch into LDS
  for (int i = tid; i < NPTS; i += QPB) {
    xs[i] = xyz[3 * i + 0];
    ys[i] = xyz[3 * i + 1];
    zs[i] = xyz[3 * i + 2];
  }

  const int n = blockIdx.x * QPB + tid;            // this lane's query index
  #pragma unroll
  for (int o = 0; o < 8; ++o) {                    // init: diag sentinel = judge, arg = self
    minv[o * QPB + tid] = judge;
    mina[o * QPB + tid] = n;
  }
  __syncthreads();

  const int wv   = tid >> 5;                       // wave in block
  const int ln   = tid & 31;                       // lane in wave
  const int half = ln >> 4;                        // 0 -> K0/K1 slots, 1 -> K2/K3
  const int mrow = ln & 15;                        // M (or N) index inside tile
  const int qb   = blockIdx.x * QPB + wv * 32;     // wave's first query
  const int sb   = wv * 512;                       // wave-private staging base

  // lane-owned query coords for the scalar (phase 2) pass
  const float lqx = xs[n], lqy = ys[n], lqz = zs[n];

  // Build A operands for the wave's two 16-row tiles (queries qb..qb+15, qb+16..qb+31)
  v2f a0, a1;
  {
    const int q0 = qb + mrow, q1 = qb + 16 + mrow;
    const float x0 = xs[q0], y0 = ys[q0], z0 = zs[q0];
    const float x1 = xs[q1], y1 = ys[q1], z1 = zs[q1];
    const float n0 = x0 * x0 + y0 * y0 + z0 * z0;
    const float n1 = x1 * x1 + y1 * y1 + z1 * z1;
    if (half == 0) { a0[0] = -2.f * x0; a0[1] = -2.f * y0; a1[0] = -2.f * x1; a1[1] = -2.f * y1; }
    else           { a0[0] = -2.f * z0; a0[1] = n0;        a1[0] = -2.f * z1; a1[1] = n1;        }
  }

  for (int c = 0; c < NCHUNK; ++c) {
    // ---- phase 1: one WMMA per 16-query tile gives the full dist^2 tile ----
    const int pcol = c * 16 + mrow;
    const float px = xs[pcol], py = ys[pcol], pz = zs[pcol];
    const float pn = px * px + py * py + pz * pz;
    v2f bb;
    if (half == 0) { bb[0] = px; bb[1] = py;  }    // K0=x, K1=y rows
    else           { bb[0] = pz; bb[1] = 1.f; }    // K2=z, K3=1 rows
    v8f cc;
    #pragma unroll
    for (int v = 0; v < 8; ++v) cc[v] = pn;        // C[m][n] = |p_n|^2

    v8f d0 = __builtin_amdgcn_wmma_f32_16x16x4_f32(false, a0, false, bb, (short)0, cc, false, false);
    v8f d1 = __builtin_amdgcn_wmma_f32_16x16x4_f32(false, a1, false, bb, (short)0, cc, false, false);

    // stage both tiles: row = M within wave's 32 queries, col = candidate j
    #pragma unroll
    for (int v = 0; v < 8; ++v) {
      dstage[sb + (v + 8 * half)      * 16 + mrow] = d0[v];
      dstage[sb + (v + 8 * half + 16) * 16 + mrow] = d1[v];
    }
    // wave-private producer->consumer through LDS: same-wave LDS is in-order,
    // s_wait_dscnt 0 guarantees completion; memory clobber stops reordering.
    asm volatile("s_wait_dscnt 0" ::: "memory");

    // ---- phase 2: lane-per-query octant argmin over the 16 candidates ----
    #pragma unroll 4
    for (int j = 0; j < 16; ++j) {
      const int pj = c * 16 + j;
      const float d  = dstage[sb + ln * 16 + j];
      const float cx = xs[pj], cy = ys[pj], cz = zs[pj];   // broadcast reads
      const int oct = ((cx >= lqx) ? 4 : 0) | ((cy >= lqy) ? 2 : 0) | ((cz >= lqz) ? 1 : 0);
      const bool valid = (pj != n) & (d > 1e-10f) & (d < judge);
      if (valid) {
        const int a = oct * QPB + tid;                     // lane-stride-1: no bank conflicts
        if (d < minv[a]) { minv[a] = d; mina[a] = pj; }    // strict < => first-min ties
      }
    }
    asm volatile("" ::: "memory");
  }

  #pragma unroll
  for (int o = 0; o < 8; ++o)
    idx_out[((size_t)(b * NPTS + n)) * 8 + o] = (float)mina[o * QPB + tid];
}

// ---------------------------------------------------------------------------
// Kernel 2: gather (bandwidth-bound; points rows sit in L2)
// ---------------------------------------------------------------------------
__global__ __launch_bounds__(256) void pointsift_gather(
    const float* __restrict__ xyz, const float* __restrict__ points,
    const float* __restrict__ idxf,
    float* __restrict__ gx,          // [B][N][8][3]
    float* __restrict__ gp)          // [B][N][8][3+CF]
{
  const int g = blockIdx.x * 256 + threadIdx.x;    // group id over B*N*8
  if (g >= NB * NPTS * 8) return;
  const int bn = g >> 3;
  const int b  = bn >> 12;                         // NPTS = 4096
  const int p  = (int)idxf[g];

  const float* q = xyz + (size_t)bn * 3;
  const float* s = xyz + ((size_t)b * NPTS + p) * 3;
  const float dx = s[0] - q[0], dy = s[1] - q[1], dz = s[2] - q[2];

  float* ox = gx + (size_t)g * 3;
  ox[0] = dx; ox[1] = dy; ox[2] = dz;

  float* op = gp + (size_t)g * (3 + CF);
  op[0] = dx; op[1] = dy; op[2] = dz;
  const float4* f = (const float4*)(points + ((size_t)b * NPTS + p) * CF);  // 256B-aligned rows
  #pragma unroll
  for (int i = 0; i < CF / 4; ++i) {
    const float4 v = f[i];
    op[3 + 4 * i + 0] = v.x; op[3 + 4 * i + 1] = v.y;
    op[3 + 4 * i + 2] = v.z; op[3 + 4 * i + 3] = v.w;
  }
}

// ---------------------------------------------------------------------------
extern "C" void kernel_launch(void* const* d_in, const int* in_sizes, int n_in,
                              void* d_out, int out_size, void* d_ws, size_t ws_size,
                              hipStream_t stream) {
  (void)in_sizes; (void)n_in; (void)out_size; (void)d_ws; (void)ws_size;

  const float* xyz    = (const float*)d_in[0];   // [B,N,3]
  const float* points = (const float*)d_in[1];   // [B,N,64]
  const float* radius = (const float*)d_in[2];   // scalar

  float* out = (float*)d_out;
  float* gx  = out;                                          // [B,N,8,3]
  float* gp  = out + (size_t)NB * NPTS * 8 * 3;              // [B,N,8,67]
  float* oix = gp  + (size_t)NB * NPTS * 8 * (3 + CF);       // [B,N,8] (as float)

  dim3 grid1(NPTS / QPB, NB);
  pointsift_select_wmma<<<grid1, QPB, 0, stream>>>(xyz, radius, oix);

  const int ngroups = NB * NPTS * 8;
  pointsift_gather<<<(ngroups + 255) / 256, 256, 0, stream>>>(xyz, points, oix, gx, gp);
}